// SelfAttentionBlock_24730421691153
// MI455X (gfx1250) — compile-verified
//
#include <hip/hip_runtime.h>
#include <math.h>

// Problem constants (match reference)
#define NDIM 64          // node feature dim
#define QKV_COLS 192     // 3 * NDIM
#define NHEAD 4
#define HEADD 16
#define ATTN_SCALE 0.25f // HEADD^-0.5

typedef float v2f __attribute__((ext_vector_type(2)));
typedef float v8f __attribute__((ext_vector_type(8)));

// ---------------------------------------------------------------------------
// Float atomic max via sign-aware int atomics (lowers to native
// global_atomic_max_i32 / global_atomic_min_u32 — no CAS loop).
// Valid because: non-negative floats order like their int bits; negative
// floats order inversely to their uint bits; any negative pattern (>=0x8000_0000
// as uint) never beats a stored positive under umin, and any positive int bits
// beat a stored negative under smax. Init value must be -inf (0xFF800000).
// ---------------------------------------------------------------------------
__device__ __forceinline__ void atomicMaxFloat(float* addr, float val) {
    if (val >= 0.0f) {
        atomicMax((int*)addr, __float_as_int(val));
    } else {
        atomicMin((unsigned int*)addr, __float_as_uint(val));
    }
}

// ---------------------------------------------------------------------------
// Kernel 0: initialize m = -inf, denom = 0, out = 0
// ---------------------------------------------------------------------------
__global__ void init_buffers_kernel(float* __restrict__ mbuf,
                                    float* __restrict__ denom,
                                    float* __restrict__ out,
                                    int nH, int nD) {
    int i = blockIdx.x * blockDim.x + threadIdx.x;
    if (i < nH) { mbuf[i] = -INFINITY; denom[i] = 0.0f; }
    if (i < nD) { out[i] = 0.0f; }
}

// ---------------------------------------------------------------------------
// Kernel 1: qkv[N,192] = x[N,64] @ w[192,64]^T + b[192]
// One wave computes one 16x16 tile of qkv with 16 chained
// V_WMMA_F32_16X16X4_F32 (K = 64, 4 per instruction).
//
// fp32 A (16x4, MxK) fragment layout (ISA 7.12.2):
//   lanes 0-15:  M=lane,    VGPR0=K(k0+0), VGPR1=K(k0+1)
//   lanes 16-31: M=lane-16, VGPR0=K(k0+2), VGPR1=K(k0+3)
// B (4x16, KxN) mirrors it with N=lane&15 (B[k][n] == w[n][k], so the load is
// a row of w, same addressing form as the A load).
// C/D (16x16 f32): VGPR r, lanes 0-15 -> M=r, N=lane; lanes 16-31 -> M=8+r.
// ---------------------------------------------------------------------------
__global__ void qkv_wmma_kernel(const float* __restrict__ x,
                                const float* __restrict__ w,
                                const float* __restrict__ bqkv,
                                float* __restrict__ qkv,
                                int n_nodes) {
    const int lane = threadIdx.x & 31;
    const int wave = threadIdx.x >> 5;                 // 0..3
    const int row_base = blockIdx.x * 16;              // node-tile
    const int col_base = (blockIdx.y * 4 + wave) * 16; // qkv-column tile (0..176)
    if (row_base + 16 > n_nodes) return;               // uniform per wave

    const int mn   = lane & 15;     // M for A, N for B/C
    const int koff = (lane >> 4) * 2;

    const float* __restrict__ xrow = x + (size_t)(row_base + mn) * NDIM;
    const float* __restrict__ wrow = w + (size_t)(col_base + mn) * NDIM;

    v8f c = {0.f, 0.f, 0.f, 0.f, 0.f, 0.f, 0.f, 0.f};
    #pragma unroll
    for (int k0 = 0; k0 < NDIM; k0 += 4) {
        v2f a, b;
        a.x = xrow[k0 + koff];
        a.y = xrow[k0 + koff + 1];
        b.x = wrow[k0 + koff];
        b.y = wrow[k0 + koff + 1];
        c = __builtin_amdgcn_wmma_f32_16x16x4_f32(
                /*neg_a=*/false, a, /*neg_b=*/false, b,
                /*c_mod=*/(short)0, c, /*reuse_a=*/false, /*reuse_b=*/false);
    }

    const float bias = bqkv[col_base + mn];
    const int mtop = (lane >> 4) * 8;
    #pragma unroll
    for (int r = 0; r < 8; ++r) {
        int row = row_base + mtop + r;
        qkv[(size_t)row * QKV_COLS + col_base + mn] = c[r] + bias;
    }
}

// ---------------------------------------------------------------------------
// Kernel 2: per-edge compatibility + segment max.
// qkv layout per node: [0:64)=Q(4 heads x 16), [64:128)=K, [128:192)=V.
// qkv table (38.4 MB) is L2-resident on MI455X (192 MB L2), so these random
// gathers stay on-chip.
// ---------------------------------------------------------------------------
__global__ void edge_compat_kernel(const float* __restrict__ qkv,
                                   const int* __restrict__ ei,
                                   float* __restrict__ compat,
                                   float* __restrict__ mbuf,
                                   int nE) {
    int e = blockIdx.x * blockDim.x + threadIdx.x;
    if (e >= nE) return;
    const int s = ei[e];        // destination segment (query)
    const int t = ei[nE + e];   // key/value node
    const float4* __restrict__ qp = (const float4*)(qkv + (size_t)s * QKV_COLS);
    const float4* __restrict__ kp = (const float4*)(qkv + (size_t)t * QKV_COLS + NDIM);
    #pragma unroll
    for (int h = 0; h < NHEAD; ++h) {
        float acc = 0.0f;
        #pragma unroll
        for (int d4 = 0; d4 < 4; ++d4) {
            float4 q = qp[h * 4 + d4];
            float4 k = kp[h * 4 + d4];
            acc += q.x * k.x + q.y * k.y + q.z * k.z + q.w * k.w;
        }
        acc *= ATTN_SCALE;
        compat[(size_t)e * NHEAD + h] = acc;
        atomicMaxFloat(&mbuf[s * NHEAD + h], acc);
    }
}

// ---------------------------------------------------------------------------
// Kernel 3: e = exp(compat - m[s]); denom[s] += e (native f32 atomic add)
// ---------------------------------------------------------------------------
__global__ void edge_exp_kernel(const int* __restrict__ ei,
                                const float* __restrict__ mbuf,
                                float* __restrict__ compat,
                                float* __restrict__ denom,
                                int nE) {
    int e = blockIdx.x * blockDim.x + threadIdx.x;
    if (e >= nE) return;
    const int s = ei[e];
    #pragma unroll
    for (int h = 0; h < NHEAD; ++h) {
        float v = __expf(compat[(size_t)e * NHEAD + h] - mbuf[s * NHEAD + h]);
        compat[(size_t)e * NHEAD + h] = v;
        unsafeAtomicAdd(&denom[s * NHEAD + h], v);
    }
}

// ---------------------------------------------------------------------------
// Kernel 4: out[s] += v[t] * (e / denom[s]); one thread per (edge, head),
// 16 native f32 atomic adds each.
// ---------------------------------------------------------------------------
__global__ void edge_agg_kernel(const float* __restrict__ qkv,
                                const int* __restrict__ ei,
                                const float* __restrict__ compat,
                                const float* __restrict__ denom,
                                float* __restrict__ out,
                                int nE) {
    int idx = blockIdx.x * blockDim.x + threadIdx.x;
    if (idx >= nE * NHEAD) return;
    const int e = idx >> 2;
    const int h = idx & 3;
    const int s = ei[e];
    const int t = ei[nE + e];
    const float attn = compat[(size_t)e * NHEAD + h] / denom[s * NHEAD + h];
    const float4* __restrict__ vp =
        (const float4*)(qkv + (size_t)t * QKV_COLS + 2 * NDIM + h * HEADD);
    float* __restrict__ op = out + (size_t)s * NDIM + h * HEADD;
    #pragma unroll
    for (int d4 = 0; d4 < 4; ++d4) {
        float4 v = vp[d4];
        unsafeAtomicAdd(op + d4 * 4 + 0, v.x * attn);
        unsafeAtomicAdd(op + d4 * 4 + 1, v.y * attn);
        unsafeAtomicAdd(op + d4 * 4 + 2, v.z * attn);
        unsafeAtomicAdd(op + d4 * 4 + 3, v.w * attn);
    }
}

// ---------------------------------------------------------------------------
// Launch: inputs are {x[N,64] f32, edge_index[2,E] int, w_qkv[192,64] f32,
// b_qkv[192] f32}; output is out[N,64] f32.
// Workspace layout: qkv (N*192 f32) | m (N*4) | denom (N*4) | compat (E*4)
//   = 38.4 MB + 0.8 + 0.8 + 16 MB ~= 56 MB.
// ---------------------------------------------------------------------------
extern "C" void kernel_launch(void* const* d_in, const int* in_sizes, int n_in,
                              void* d_out, int out_size, void* d_ws, size_t ws_size,
                              hipStream_t stream) {
    const float* x    = (const float*)d_in[0];
    const int*   ei   = (const int*)d_in[1];
    const float* wqkv = (const float*)d_in[2];
    const float* bqkv = (const float*)d_in[3];
    float*       out  = (float*)d_out;

    const int nN = in_sizes[0] / NDIM;   // 50000
    const int nE = in_sizes[1] / 2;      // 1000000

    float* qkv    = (float*)d_ws;
    float* mbuf   = qkv   + (size_t)nN * QKV_COLS;
    float* denom  = mbuf  + (size_t)nN * NHEAD;
    float* compat = denom + (size_t)nN * NHEAD;

    const int nH = nN * NHEAD;
    const int nD = nN * NDIM;

    init_buffers_kernel<<<(nD + 255) / 256, 256, 0, stream>>>(mbuf, denom, out, nH, nD);

    // 3125 row tiles x 12 col tiles; 4 waves (4 col tiles) per 128-thread block
    dim3 ggrid((nN + 15) / 16, (QKV_COLS / 16) / 4);
    qkv_wmma_kernel<<<ggrid, 128, 0, stream>>>(x, wqkv, bqkv, qkv, nN);

    edge_compat_kernel<<<(nE + 255) / 256, 256, 0, stream>>>(qkv, ei, compat, mbuf, nE);
    edge_exp_kernel<<<(nE + 255) / 256, 256, 0, stream>>>(ei, mbuf, compat, denom, nE);
    edge_agg_kernel<<<(nE * NHEAD + 255) / 256, 256, 0, stream>>>(qkv, ei, compat, denom, out, nE);
}